// Depth3DGridGen_58308476010672
// MI455X (gfx1250) — compile-verified
//
#include <hip/hip_runtime.h>
#include <math.h>

typedef __attribute__((ext_vector_type(2))) float v2f;
typedef __attribute__((ext_vector_type(8))) float v8f;

#define HEIGHT 1024
#define WIDTH  2048
#define NPIX   (HEIGHT * WIDTH)
#define PI_F   3.14159265358979f

// lane <-> lane^16 swap via ds_swizzle SWAPX16 (group-of-32: and=0x1f, or=0, xor=0x10).
// No address VGPR needed (unlike __shfl_xor -> lshlrev + ds_bpermute).
// Must be executed with full EXEC (reads from disabled lanes return 0).
__device__ __forceinline__ float swz16(float v) {
    int i = __builtin_bit_cast(int, v);
    i = __builtin_amdgcn_ds_swizzle(i, 0x401f);
    return __builtin_bit_cast(float, i);
}

// acos(x) ~ sqrt(1-|x|)*poly(|x|), reflected for x<0 (abs err ~7e-5 rad)
__device__ __forceinline__ float fast_acos(float x) {
    float xa = fabsf(x);
    float t  = __builtin_amdgcn_sqrtf(fmaxf(1.0f - xa, 0.0f));
    float p  = fmaf(xa, -0.0187293f, 0.0742610f);
    p = fmaf(xa, p, -0.2121144f);
    p = fmaf(xa, p,  1.5707288f);
    float r = t * p;
    return (x < 0.0f) ? (PI_F - r) : r;
}

// atan2 via min/max-ratio polynomial + quadrant fixups
__device__ __forceinline__ float fast_atan2(float y, float x) {
    float ax = fabsf(x), ay = fabsf(y);
    float mx = fmaxf(ax, ay);
    float mn = fminf(ax, ay);
    float a  = mn * __builtin_amdgcn_rcpf(fmaxf(mx, 1e-30f));
    float s  = a * a;
    float r  = fmaf(s, 0.0208351f, -0.0851330f);
    r = fmaf(s, r,  0.1801410f);
    r = fmaf(s, r, -0.3302995f);
    r = fmaf(s, r,  0.9998660f);
    r = r * a;
    if (ay > ax)   r = 1.57079632679f - r;
    if (x < 0.0f)  r = PI_F - r;
    return copysignf(r, y);
}

__global__ __launch_bounds__(256) void Depth3DGridGen_kernel(
    const float* __restrict__ depth,       // (4, H, W, 1)
    const float* __restrict__ transform,   // (4, 4, 4)
    float* __restrict__ out)               // (4, H, W, 2)
{
    const int tid  = blockIdx.x * blockDim.x + threadIdx.x;   // one (h,w) per thread
    const int lane = threadIdx.x & 31;
    const int w = tid & (WIDTH - 1);
    const int h = tid >> 11;

    // grid direction (amortized over the 4 batches)
    const float theta = PI_F * (float)h * (1.0f / (float)HEIGHT);
    const float phi   = PI_F * fmaf((float)w, 2.0f / (float)WIDTH, -1.0f);
    float st, ct, sp, cp;
    __sincosf(theta, &st, &ct);
    __sincosf(phi,   &sp, &cp);
    const float gx = st * cp;
    const float gy = st * sp;
    const float gz = ct;

    // A = T^T fragments in the 16x4 f32 A layout:
    //   v0: lanes 0-15 -> A[M=lane][K=0], lanes 16-31 -> A[M=lane-16][K=2]
    //   v1:               K=1                              K=3
    // A[M=j][K=k] = T[b][k][j]; columns j>3 clamp to 3 (finite, unused in D)
    const int jc   = (lane & 15) < 3 ? (lane & 15) : 3;
    const int krow = (lane < 16) ? 0 : 2;
    v2f A[4];
#pragma unroll
    for (int b = 0; b < 4; ++b) {
        const float* Tb = transform + b * 16;
        A[b].x = Tb[(krow + 0) * 4 + jc];
        A[b].y = Tb[(krow + 1) * 4 + jc];
    }

    const bool lo = (lane < 16);

#pragma unroll
    for (int b = 0; b < 4; ++b) {
        const float d  = depth[b * NPIX + tid];
        const float px = gx * d;
        const float py = gy * d;
        const float pz = gz * d;

        // unconditional cross-half copies (full EXEC) to assemble B fragments
        const float pxs = swz16(px);
        const float pys = swz16(py);
        const float pzs = swz16(pz);

        // B layout (mirror of A): v0 = rows K=0 (lanes 0-15) / K=2 (16-31); v1 = K=1 / K=3
        v2f B1, B2;
        B1.x = lo ? px  : pzs;    // pixels 0-15
        B1.y = lo ? py  : 1.0f;
        B2.x = lo ? pxs : pz;     // pixels 16-31
        B2.y = lo ? pys : 1.0f;

        v8f C = {};
        // D = (T^T)(4x4 in 16x4) x points(4x16): rows 0,1,2 of D = X,Y,Z per pixel-lane
        v8f D1 = __builtin_amdgcn_wmma_f32_16x16x4_f32(
            false, A[b], false, B1, (short)0, C, false, false);
        v8f D2 = __builtin_amdgcn_wmma_f32_16x16x4_f32(
            false, A[b], false, B2, (short)0, C, false, false);

        // Hoisted, unconditional swizzles of D2 rows 0-2 (so no exec-mask branching),
        // then merge: lanes 0-15 own pixels 0-15 (D1), lanes 16-31 pixels 16-31 (D2).
        const float d2x = swz16(D2[0]);
        const float d2y = swz16(D2[1]);
        const float d2z = swz16(D2[2]);
        const float X = lo ? D1[0] : d2x;
        const float Y = lo ? D1[1] : d2y;
        const float Z = lo ? D1[2] : d2z;

        const float r  = __builtin_amdgcn_sqrtf(fmaf(X, X, fmaf(Y, Y, Z * Z))) + 1e-4f;
        const float th = fast_acos(Z * __builtin_amdgcn_rcpf(r)) * (2.0f / PI_F) - 1.0f;
        const float ph = fast_atan2(Y, X) * (1.0f / PI_F);

        float2 o = make_float2(ph, th);
        *(float2*)(out + (size_t)(b * NPIX + tid) * 2) = o;   // coalesced b64 store
    }
}

extern "C" void kernel_launch(void* const* d_in, const int* in_sizes, int n_in,
                              void* d_out, int out_size, void* d_ws, size_t ws_size,
                              hipStream_t stream) {
    const float* depth     = (const float*)d_in[0];   // (4,1024,2048,1) f32
    const float* transform = (const float*)d_in[1];   // (4,4,4) f32
    float* out             = (float*)d_out;           // (4,1024,2048,2) f32

    dim3 block(256);
    dim3 grid(NPIX / 256);  // 8192 blocks, grid exactly covers all pixels (EXEC all-ones)
    hipLaunchKernelGGL(Depth3DGridGen_kernel, grid, block, 0, stream,
                       depth, transform, out);
}